// SimpleGCN_4389456577426
// MI455X (gfx1250) — compile-verified
//
#include <hip/hip_runtime.h>
#include <hip/hip_bf16.h>

typedef __attribute__((ext_vector_type(2))) float v2f;
typedef __attribute__((ext_vector_type(8))) float v8f;

#define NCH 64

// ---------------- degree / normalization ----------------
__global__ void deg_init_kernel(float* deg, int n) {
    int i = blockIdx.x * blockDim.x + threadIdx.x;
    if (i < n) deg[i] = 1.0f;  // self-loop
}

__global__ void deg_count_kernel(const int* __restrict__ dst, float* deg, int e) {
    int i = blockIdx.x * blockDim.x + threadIdx.x;
    if (i < e) atomicAdd(&deg[dst[i]], 1.0f);
}

__global__ void deg_rsqrt_kernel(float* deg, int n) {
    int i = blockIdx.x * blockDim.x + threadIdx.x;
    if (i < n) deg[i] = rsqrtf(deg[i]);
}

__global__ void zero_kernel(float* p, int n) {
    int i = blockIdx.x * blockDim.x + threadIdx.x;
    if (i < n) p[i] = 0.0f;
}

// ---------------- dense GEMM via V_WMMA_F32_16X16X4_F32 ----------------
// H[n x 64] = X[n x 64] @ W[64 x 64].  One wave per 16x16 output tile,
// 4 waves/block cover the 4 N-tiles of the 64 output channels.
// n must be a multiple of 16 (50000 = 3125*16). No divergence: EXEC all-1s.
__global__ __launch_bounds__(128) void gcn_gemm64_wmma(
    const float* __restrict__ X, const float* __restrict__ W,
    float* __restrict__ H, int n) {
    const int lane = threadIdx.x & 31;
    const int wave = threadIdx.x >> 5;       // N tile: 0..3
    const int half = lane >> 4;              // 0 or 1 (lane group)
    const int l    = lane & 15;
    const int row0 = blockIdx.x * 16;
    const int col0 = wave * 16;

    v8f c = {};
    #pragma unroll
    for (int k = 0; k < NCH; k += 4) {
        // A 16x4 (MxK): lanes 0-15 -> {K=k, k+1}, lanes 16-31 -> {K=k+2, k+3}
        const float* xr = X + (size_t)(row0 + l) * NCH + k + half * 2;
        v2f a; a.x = xr[0]; a.y = xr[1];
        // B 4x16 (KxN): lanes 0-15 -> {K=k, k+1}, lanes 16-31 -> {K=k+2, k+3}
        const float* wr = W + (size_t)(k + half * 2) * NCH + col0 + l;
        v2f b; b.x = wr[0]; b.y = wr[NCH];
        c = __builtin_amdgcn_wmma_f32_16x16x4_f32(
            /*neg_a=*/false, a, /*neg_b=*/false, b,
            /*c_mod=*/(short)0, c, /*reuse_a=*/false, /*reuse_b=*/false);
    }
    // D 16x16: VGPR v -> rows (v, v+8) split across lane halves
    #pragma unroll
    for (int v = 0; v < 8; ++v) {
        int row = row0 + v + half * 8;
        H[(size_t)row * NCH + col0 + l] = c[v];
    }
}

// ---------------- edge scatter: acc[dst] += dinv[src]*dinv[dst]*h[src] ----------------
// 64 lanes per edge (one lane per channel): coalesced gather + f32 global atomics.
__global__ __launch_bounds__(256) void scatter_edges_kernel(
    const int* __restrict__ src, const int* __restrict__ dst,
    const float* __restrict__ dinv, const float* __restrict__ h,
    float* __restrict__ acc, int e) {
    int eid = blockIdx.x * 4 + (threadIdx.x >> 6);
    if (eid >= e) return;
    int c = threadIdx.x & 63;
    int s = src[eid];
    int d = dst[eid];
    float norm = dinv[s] * dinv[d];
    atomicAdd(&acc[(size_t)d * NCH + c], norm * h[(size_t)s * NCH + c]);
}

// ---------------- fuse self-loop + bias + ReLU (in place on acc) ----------------
__global__ void finish_layer_kernel(const float* __restrict__ h,
                                    const float* __restrict__ dinv,
                                    const float* __restrict__ bias,
                                    float* __restrict__ acc, int total) {
    int i = blockIdx.x * blockDim.x + threadIdx.x;
    if (i >= total) return;
    int node = i >> 6;
    int c = i & 63;
    float di = dinv[node];
    float v = acc[i] + di * di * h[i] + bias[c];
    acc[i] = v > 0.0f ? v : 0.0f;
}

// ---------------- global mean pool (segment sums) ----------------
__global__ __launch_bounds__(256) void pool_accum_kernel(
    const float* __restrict__ h, const int* __restrict__ batch,
    float* __restrict__ sums, float* __restrict__ cnt, int n) {
    int node = blockIdx.x * 4 + (threadIdx.x >> 6);
    if (node >= n) return;
    int c = threadIdx.x & 63;
    int g = batch[node];
    atomicAdd(&sums[(size_t)g * NCH + c], h[(size_t)node * NCH + c]);
    if (c == 0) atomicAdd(&cnt[g], 1.0f);
}

__global__ void pool_final_kernel(const float* __restrict__ sums,
                                  const float* __restrict__ cnt,
                                  const float* __restrict__ Wl,
                                  const float* __restrict__ bl,
                                  float* __restrict__ out, int ngraphs) {
    int g = blockIdx.x * blockDim.x + threadIdx.x;
    if (g >= ngraphs) return;
    float inv = 1.0f / fmaxf(cnt[g], 1.0f);
    float acc = 0.0f;
    #pragma unroll
    for (int c = 0; c < NCH; ++c) acc += sums[(size_t)g * NCH + c] * Wl[c];
    out[g] = acc * inv + bl[0];
}

extern "C" void kernel_launch(void* const* d_in, const int* in_sizes, int n_in,
                              void* d_out, int out_size, void* d_ws, size_t ws_size,
                              hipStream_t stream) {
    const float* x     = (const float*)d_in[0];
    const int*   ei    = (const int*)d_in[1];
    const int*   batch = (const int*)d_in[2];
    const float* W1    = (const float*)d_in[3];
    const float* b1    = (const float*)d_in[4];
    const float* W2    = (const float*)d_in[5];
    const float* b2    = (const float*)d_in[6];
    const float* Wl    = (const float*)d_in[7];
    const float* bl    = (const float*)d_in[8];
    float* out = (float*)d_out;

    const int N = in_sizes[0] / NCH;     // 50000
    const int E = in_sizes[1] / 2;       // 800000
    const int G = out_size;              // 256
    const int NF = N * NCH;

    const int* src = ei;
    const int* dst = ei + E;

    // Workspace layout (floats)
    float* ws   = (float*)d_ws;
    float* dinv = ws;                                  // N (padded)
    float* bufA = dinv + ((N + 63) & ~63);             // N*64
    float* bufB = bufA + (size_t)NF;                   // N*64
    float* sums = bufB + (size_t)NF;                   // G*64
    float* cnt  = sums + (size_t)G * NCH;              // G

    const int T = 256;
    // 1) symmetric-normalization coefficients
    deg_init_kernel<<<(N + T - 1) / T, T, 0, stream>>>(dinv, N);
    deg_count_kernel<<<(E + T - 1) / T, T, 0, stream>>>(dst, dinv, E);
    deg_rsqrt_kernel<<<(N + T - 1) / T, T, 0, stream>>>(dinv, N);

    // 2) layer 1: H1 = X @ W1 (WMMA), scatter, self-loop+bias+ReLU
    gcn_gemm64_wmma<<<N / 16, 128, 0, stream>>>(x, W1, bufA, N);
    zero_kernel<<<(NF + T - 1) / T, T, 0, stream>>>(bufB, NF);
    scatter_edges_kernel<<<(E + 3) / 4, 256, 0, stream>>>(src, dst, dinv, bufA, bufB, E);
    finish_layer_kernel<<<(NF + T - 1) / T, T, 0, stream>>>(bufA, dinv, b1, bufB, NF);
    // bufB now holds layer-1 activations

    // 3) layer 2: H2 = X1 @ W2 (WMMA) into bufA, then reuse bufB as accumulator
    gcn_gemm64_wmma<<<N / 16, 128, 0, stream>>>(bufB, W2, bufA, N);
    zero_kernel<<<(NF + T - 1) / T, T, 0, stream>>>(bufB, NF);
    scatter_edges_kernel<<<(E + 3) / 4, 256, 0, stream>>>(src, dst, dinv, bufA, bufB, E);
    finish_layer_kernel<<<(NF + T - 1) / T, T, 0, stream>>>(bufA, dinv, b2, bufB, NF);
    // bufB now holds layer-2 activations

    // 4) global mean pool + readout
    zero_kernel<<<(G * NCH + T - 1) / T, T, 0, stream>>>(sums, G * NCH);
    zero_kernel<<<(G + T - 1) / T, T, 0, stream>>>(cnt, G);
    pool_accum_kernel<<<(N + 3) / 4, 256, 0, stream>>>(bufB, batch, sums, cnt, N);
    pool_final_kernel<<<1, 256, 0, stream>>>(sums, cnt, Wl, bl, out, G);
}